// DECOLA_Stage1Assigner_67456756351519
// MI455X (gfx1250) — compile-verified
//
#include <hip/hip_runtime.h>
#include <hip/hip_bf16.h>
#include <stdint.h>

// ---------------------------------------------------------------------------
// DECOLA Stage-1 assigner for MI455X (gfx1250, wave32).
// B=8 images, N=20000 anchors, G=300 gts, C=80 classes, K=4, MAX_POS=128.
// Latency/VALU bound (~3MB inputs); the only bilinear term of the IoU
// (area_gt + area_anchor outer-sum in the union) runs on the matrix pipe via
// V_WMMA_F32_16X16X4_F32 (rank-2: A=[area_g,1], B=[1;area_n]). gt staging in
// k_amax uses GLOBAL_LOAD_ASYNC_TO_LDS_B32 + s_wait_asynccnt (ASYNCcnt path).
// Per-gt top-K is split 10x over the anchor range for wave occupancy and
// merged with lax.top_k's (value desc, index asc) stable order.
// ---------------------------------------------------------------------------

#define DEV __device__ __forceinline__

constexpr int   B_     = 8;
constexpr int   N_     = 20000;
constexpr int   G_     = 300;
constexpr int   C_     = 80;
constexpr int   K_     = 4;
constexpr int   MAXPOS = 128;
constexpr float T_HIGH = 0.7f;
constexpr int   GTILES = (G_ + 15) / 16;  // 19
constexpr int   NTILES = N_ / 16;         // 1250 (exact)
constexpr int   SPLIT  = 10;              // anchor-range split for k_topk
constexpr int   TPS    = NTILES / SPLIT;  // 125 tiles per split (exact)

typedef float v2f __attribute__((ext_vector_type(2)));
typedef float v8f __attribute__((ext_vector_type(8)));

// Contraction OFF so every kernel produces bit-identical IoU values: the
// reference tests iou == gt_max with exact float equality across passes.
DEV void cxcywh_xyxy_area(float cx, float cy, float w, float h,
                          float& x0, float& y0, float& x1, float& y1, float& ar) {
#pragma clang fp contract(off)
  x0 = cx - 0.5f * w; y0 = cy - 0.5f * h;
  x1 = cx + 0.5f * w; y1 = cy + 0.5f * h;
  ar = (x1 - x0) * (y1 - y0);
}

DEV float fadd_off(float a, float b) {
#pragma clang fp contract(off)
  return a + b;
}

// gt box first, anchor box second, everywhere (identical op order => identical bits)
DEV float iou_from_sum(float ax0, float ay0, float ax1, float ay1,
                       float bx0, float by0, float bx1, float by1, float areasum) {
#pragma clang fp contract(off)
  float ix0 = fmaxf(ax0, bx0), iy0 = fmaxf(ay0, by0);
  float ix1 = fminf(ax1, bx1), iy1 = fminf(ay1, by1);
  float iw = fmaxf(ix1 - ix0, 0.0f), ih = fmaxf(iy1 - iy0, 0.0f);
  float inter = iw * ih;
  return inter / (areasum - inter);
}

// ---------------- gt prep: cxcywh -> xyxy + area, AoS[5] --------------------
__global__ void k_gtprep(const float* __restrict__ tb, float* __restrict__ gtbox) {
  int i = blockIdx.x * blockDim.x + threadIdx.x;
  if (i >= B_ * G_) return;
  float x0, y0, x1, y1, ar;
  cxcywh_xyxy_area(tb[i * 4 + 0], tb[i * 4 + 1], tb[i * 4 + 2], tb[i * 4 + 3],
                   x0, y0, x1, y1, ar);
  gtbox[i * 5 + 0] = x0; gtbox[i * 5 + 1] = y0;
  gtbox[i * 5 + 2] = x1; gtbox[i * 5 + 3] = y1;
  gtbox[i * 5 + 4] = ar;
}

// ---------------- per-anchor max / argmax over class-masked gts -------------
// gt data staged into LDS with the gfx1250 async-to-LDS engine (ASYNCcnt).
__global__ __launch_bounds__(256) void k_amax(const float* __restrict__ anchors,
                                              const int* __restrict__ prompt,
                                              const int* __restrict__ tlab,
                                              const float* __restrict__ gtbox,
                                              float* __restrict__ amax,
                                              int* __restrict__ agt) {
  // single LDS object -> wave LDS_BASE-relative byte offsets start at 0
  __shared__ __align__(16) float sbuf[G_ * 6];  // [0,1500): box AoS5, [1500,1800): labels
  int b = blockIdx.y, tid = threadIdx.x;
  const float* gsrc = gtbox + (size_t)b * G_ * 5;
  const int*   lsrc = tlab + (size_t)b * G_;
  for (int i = tid; i < G_ * 5; i += 256) {
    unsigned lds_off = (unsigned)(i * 4);
    unsigned long long ga = (unsigned long long)(const void*)(gsrc + i);
    asm volatile("global_load_async_to_lds_b32 %0, %1, off"
                 :: "v"(lds_off), "v"(ga) : "memory");
  }
  for (int i = tid; i < G_; i += 256) {
    unsigned lds_off = (unsigned)((G_ * 5 + i) * 4);
    unsigned long long ga = (unsigned long long)(const void*)(lsrc + i);
    asm volatile("global_load_async_to_lds_b32 %0, %1, off"
                 :: "v"(lds_off), "v"(ga) : "memory");
  }
  asm volatile("s_wait_asynccnt 0" ::: "memory");
  __syncthreads();
  const float* sg = sbuf;
  const int*   sl = (const int*)&sbuf[G_ * 5];

  int n = blockIdx.x * 256 + tid;
  if (n >= N_) return;
  float4 a = reinterpret_cast<const float4*>(anchors)[(size_t)b * N_ + n];
  float ax0, ay0, ax1, ay1, aar;
  cxcywh_xyxy_area(a.x, a.y, a.z, a.w, ax0, ay0, ax1, ay1, aar);
  int cls = prompt[(size_t)b * N_ + n];
  float best = -1.0f;  // matches iou_m fill value; argmax of all -1 row -> 0
  int bg = 0;
  for (int g = 0; g < G_; ++g) {
    if (sl[g] != cls) continue;
    const float* p = &sg[g * 5];
    float v = iou_from_sum(p[0], p[1], p[2], p[3], ax0, ay0, ax1, ay1,
                           fadd_off(p[4], aar));
    if (v > best) { best = v; bg = g; }  // strict > keeps first occurrence
  }
  amax[(size_t)b * N_ + n] = best;
  agt[(size_t)b * N_ + n]  = bg;
}

// ------- per-gt partial top-K, WMMA-tiled, split over anchor range ----------
__global__ __launch_bounds__(32) void k_topk(const float* __restrict__ anchors,
                                             const int* __restrict__ prompt,
                                             const int* __restrict__ tlab,
                                             const float* __restrict__ gtbox,
                                             float* __restrict__ ptkv,
                                             int* __restrict__ ptki) {
  __shared__ float gx0[16], gy0[16], gx1[16], gy1[16], gar[16];
  __shared__ int   glab[16];
  __shared__ float lv[16 * 64];
  __shared__ int   li[16 * 64];

  int b = blockIdx.y;
  int tile0 = blockIdx.x * 16;
  int sp = blockIdx.z;
  int l = threadIdx.x;
  if (l < 16) {
    int g = tile0 + l;
    if (g < G_) {
      const float* p = &gtbox[((size_t)b * G_ + g) * 5];
      gx0[l] = p[0]; gy0[l] = p[1]; gx1[l] = p[2]; gy1[l] = p[3]; gar[l] = p[4];
      glab[l] = tlab[b * G_ + g];
    } else {
      gx0[l] = 0.f; gy0[l] = 0.f; gx1[l] = 0.f; gy1[l] = 0.f; gar[l] = 0.f;
      glab[l] = -2;  // never equals a prompt class (>=0)
    }
  }
  __syncthreads();

  const bool lo = l < 16;
  const int  col = l & 15;

  float tv[8][4];
  int   ti[8][4];
#pragma unroll
  for (int j = 0; j < 8; ++j)
#pragma unroll
    for (int s = 0; s < 4; ++s) { tv[j][s] = -3.0e38f; ti[j][s] = 0x7fffffff; }

  // A (16x4 f32): lanes0-15 hold K0 (gt area) and K1 (1.0); lanes16-31 hold
  // K2/K3 which we zero, so the B lane-half ordering ambiguity cancels.
  v2f amat;
  amat.x = lo ? gar[col] : 0.0f;
  amat.y = lo ? 1.0f : 0.0f;

  const float4* a4 = reinterpret_cast<const float4*>(anchors) + (size_t)b * N_;
  const int*    pr = prompt + (size_t)b * N_;

  for (int t = sp * TPS; t < (sp + 1) * TPS; ++t) {
    int n = t * 16 + col;
    float4 ab = a4[n];
    __builtin_prefetch((const void*)(a4 + n + 16), 0, 0);  // next tile -> global_prefetch_b8
    float ax0, ay0, ax1, ay1, aar;
    cxcywh_xyxy_area(ab.x, ab.y, ab.z, ab.w, ax0, ay0, ax1, ay1, aar);
    int cls = pr[n];

    // B (4x16 f32), hedged so D = gt_area + anchor_area under either
    // lane-half K-row ordering ({K0,K2}/{K1,K3} or {K0,K1}/{K2,K3}):
    v2f bmat;
    bmat.x = lo ? 1.0f : aar;
    bmat.y = lo ? aar : 0.0f;

    v8f c = {};
    c = __builtin_amdgcn_wmma_f32_16x16x4_f32(false, amat, false, bmat,
                                              (short)0, c, false, false);
    float sums[8];
#pragma unroll
    for (int j = 0; j < 8; ++j) sums[j] = c[j];

#pragma unroll
    for (int j = 0; j < 8; ++j) {
      int m = lo ? j : (j + 8);
      float v = (glab[m] == cls)
                    ? iou_from_sum(gx0[m], gy0[m], gx1[m], gy1[m],
                                   ax0, ay0, ax1, ay1, sums[j])
                    : -1.0f;
      // sorted insert, strict > keeps earliest anchor index on ties
      if (v > tv[j][3]) {
        if (v > tv[j][2]) {
          tv[j][3] = tv[j][2]; ti[j][3] = ti[j][2];
          if (v > tv[j][1]) {
            tv[j][2] = tv[j][1]; ti[j][2] = ti[j][1];
            if (v > tv[j][0]) {
              tv[j][1] = tv[j][0]; ti[j][1] = ti[j][0];
              tv[j][0] = v; ti[j][0] = n;
            } else { tv[j][1] = v; ti[j][1] = n; }
          } else { tv[j][2] = v; ti[j][2] = n; }
        } else { tv[j][3] = v; ti[j][3] = n; }
      }
    }
  }

#pragma unroll
  for (int j = 0; j < 8; ++j) {
    int row = lo ? j : (j + 8);
    int base = row * 64 + col * 4;
#pragma unroll
    for (int s = 0; s < 4; ++s) { lv[base + s] = tv[j][s]; li[base + s] = ti[j][s]; }
  }
  __syncthreads();

  if (l < 16) {
    int g = tile0 + l;
    if (g < G_) {
      // each lane saw TPS(=125) >= 4 candidates per row: all slots real (>= -1)
      uint64_t used = 0;
      for (int k = 0; k < K_; ++k) {
        float bv = -3.2e38f; int bi = 0x7fffffff; int bs = 0;
        for (int s = 0; s < 64; ++s) {
          if ((used >> s) & 1ull) continue;
          float v = lv[l * 64 + s];
          int   i = li[l * 64 + s];
          if (v > bv || (v == bv && i < bi)) { bv = v; bi = i; bs = s; }
        }
        used |= (1ull << bs);
        size_t o = ((size_t)(b * G_ + g) * SPLIT + sp) * K_ + k;
        ptkv[o] = bv;
        ptki[o] = bi;
      }
    }
  }
}

// ---------------- merge split partials -> global top-K + gt max -------------
__global__ void k_tmerge(const float* __restrict__ ptkv, const int* __restrict__ ptki,
                         float* __restrict__ tkv, int* __restrict__ tki,
                         float* __restrict__ gmax) {
  int bg = blockIdx.x * blockDim.x + threadIdx.x;
  if (bg >= B_ * G_) return;
  const float* pv = ptkv + (size_t)bg * SPLIT * K_;
  const int*   pi = ptki + (size_t)bg * SPLIT * K_;
  uint64_t used = 0;  // SPLIT*K_ = 40 <= 64
  for (int k = 0; k < K_; ++k) {
    float bv = -3.2e38f; int bi = 0x7fffffff; int bs = 0;
    for (int s = 0; s < SPLIT * K_; ++s) {
      if ((used >> s) & 1ull) continue;
      float v = pv[s]; int i = pi[s];
      if (v > bv || (v == bv && i < bi)) { bv = v; bi = i; bs = s; }
    }
    used |= (1ull << bs);
    tkv[(size_t)bg * K_ + k] = bv;
    tki[(size_t)bg * K_ + k] = bi;
    if (k == 0) gmax[bg] = bv;
  }
}

// ---------------- positives: threshold OR low-quality (iou == gt_max) -------
__global__ __launch_bounds__(256) void k_pos(const float* __restrict__ anchors,
                                             const int* __restrict__ prompt,
                                             const int* __restrict__ tlab,
                                             const float* __restrict__ gtbox,
                                             const float* __restrict__ gmax,
                                             const float* __restrict__ amax,
                                             int* __restrict__ pos) {
  __shared__ float sg[G_ * 5];
  __shared__ int   sl[G_];
  __shared__ float sm[G_];
  int b = blockIdx.y, tid = threadIdx.x;
  for (int i = tid; i < G_ * 5; i += 256) sg[i] = gtbox[b * G_ * 5 + i];
  for (int i = tid; i < G_; i += 256) { sl[i] = tlab[b * G_ + i]; sm[i] = gmax[b * G_ + i]; }
  __syncthreads();
  int n = blockIdx.x * 256 + tid;
  if (n >= N_) return;
  float4 a = reinterpret_cast<const float4*>(anchors)[(size_t)b * N_ + n];
  float ax0, ay0, ax1, ay1, aar;
  cxcywh_xyxy_area(a.x, a.y, a.z, a.w, ax0, ay0, ax1, ay1, aar);
  int cls = prompt[(size_t)b * N_ + n];
  bool p = amax[(size_t)b * N_ + n] >= T_HIGH;
  if (!p) {
    for (int g = 0; g < G_; ++g) {
      if (sl[g] != cls) continue;
      float gm = sm[g];
      if (gm < 0.0f) continue;
      const float* q = &sg[g * 5];
      float v = iou_from_sum(q[0], q[1], q[2], q[3], ax0, ay0, ax1, ay1,
                             fadd_off(q[4], aar));
      if (v == gm) { p = true; break; }
    }
  }
  pos[(size_t)b * N_ + n] = p ? 1 : 0;
}

// ---------------- first MAX_POS positives per (image,class), anchor order ---
__global__ __launch_bounds__(32) void k_sub(const int* __restrict__ prompt,
                                            const int* __restrict__ pos,
                                            int* __restrict__ pos2) {
  int bc = blockIdx.x;
  int b = bc / C_, c = bc % C_;
  int lane = threadIdx.x;  // wave32
  const int* pr = prompt + (size_t)b * N_;
  const int* po = pos + (size_t)b * N_;
  int*       p2 = pos2 + (size_t)b * N_;
  unsigned lmask = (1u << lane) - 1u;
  int count = 0;
  for (int base = 0; base < N_; base += 32) {  // N_ % 32 == 0
    int n = base + lane;
    bool mine = (pr[n] == c);
    bool p = mine && (po[n] != 0);
    unsigned m = (unsigned)__ballot(p);  // low 32 bits on wave32
    int rank = count + __popc(m & lmask);
    if (mine) p2[n] = (p && rank < MAXPOS) ? 1 : 0;  // exactly one class owns n
    count += __popc(m);
  }
}

__global__ void k_zero(int* __restrict__ cnt) {
  int i = blockIdx.x * blockDim.x + threadIdx.x;
  if (i < B_ * G_) cnt[i] = 0;
}

__global__ __launch_bounds__(256) void k_cnt(const int* __restrict__ pos2,
                                             const int* __restrict__ agt,
                                             int* __restrict__ cnt) {
  int b = blockIdx.y;
  int n = blockIdx.x * 256 + threadIdx.x;
  if (n >= N_) return;
  if (pos2[(size_t)b * N_ + n])
    atomicAdd(&cnt[b * G_ + agt[(size_t)b * N_ + n]], 1);
}

// ---------------- emit (pr, gt, valid, pos_iou) concatenated as float -------
__global__ void k_final(const float* __restrict__ tkv, const int* __restrict__ tki,
                        const int* __restrict__ cnt, float* __restrict__ out) {
  int idx = blockIdx.x * blockDim.x + threadIdx.x;
  const int BGK = B_ * G_ * K_;
  if (idx >= BGK) return;
  int k = idx % K_;
  int bg = idx / K_;
  int g = bg % G_;
  int c = cnt[bg];
  bool valid = k < (c < K_ ? c : K_);
  out[idx]           = valid ? (float)tki[idx] : -1.0f;
  out[BGK + idx]     = valid ? (float)g : -1.0f;
  out[2 * BGK + idx] = valid ? 1.0f : 0.0f;
  out[3 * BGK + idx] = valid ? tkv[idx] : 0.0f;
}

extern "C" void kernel_launch(void* const* d_in, const int* in_sizes, int n_in,
                              void* d_out, int out_size, void* d_ws, size_t ws_size,
                              hipStream_t stream) {
  const float* anchors   = (const float*)d_in[2];
  const int*   prompt    = (const int*)d_in[3];
  const int*   tgt_label = (const int*)d_in[4];
  const float* tgt_boxes = (const float*)d_in[5];

  // workspace carve-out (~3.5 MB total)
  char* w = (char*)d_ws;
  float* gtbox = (float*)w; w += (size_t)B_ * G_ * 5 * 4;
  float* amax  = (float*)w; w += (size_t)B_ * N_ * 4;
  int*   agt   = (int*)w;   w += (size_t)B_ * N_ * 4;
  int*   pos   = (int*)w;   w += (size_t)B_ * N_ * 4;
  int*   pos2  = (int*)w;   w += (size_t)B_ * N_ * 4;
  float* ptkv  = (float*)w; w += (size_t)B_ * G_ * SPLIT * K_ * 4;
  int*   ptki  = (int*)w;   w += (size_t)B_ * G_ * SPLIT * K_ * 4;
  float* tkv   = (float*)w; w += (size_t)B_ * G_ * K_ * 4;
  int*   tki   = (int*)w;   w += (size_t)B_ * G_ * K_ * 4;
  float* gmax  = (float*)w; w += (size_t)B_ * G_ * 4;
  int*   cnt   = (int*)w;   w += (size_t)B_ * G_ * 4;

  dim3 grdN((N_ + 255) / 256, B_);

  k_gtprep<<<(B_ * G_ + 255) / 256, 256, 0, stream>>>(tgt_boxes, gtbox);
  k_amax<<<grdN, 256, 0, stream>>>(anchors, prompt, tgt_label, gtbox, amax, agt);
  k_topk<<<dim3(GTILES, B_, SPLIT), 32, 0, stream>>>(anchors, prompt, tgt_label,
                                                     gtbox, ptkv, ptki);
  k_tmerge<<<(B_ * G_ + 255) / 256, 256, 0, stream>>>(ptkv, ptki, tkv, tki, gmax);
  k_pos<<<grdN, 256, 0, stream>>>(anchors, prompt, tgt_label, gtbox, gmax, amax, pos);
  k_sub<<<B_ * C_, 32, 0, stream>>>(prompt, pos, pos2);
  k_zero<<<(B_ * G_ + 255) / 256, 256, 0, stream>>>(cnt);
  k_cnt<<<grdN, 256, 0, stream>>>(pos2, agt, cnt);
  k_final<<<(B_ * G_ * K_ + 255) / 256, 256, 0, stream>>>(tkv, tki, cnt, (float*)d_out);
}